// TopkRouter_53721450939141
// MI455X (gfx1250) — compile-verified
//
#include <hip/hip_runtime.h>

typedef float v2f __attribute__((ext_vector_type(2)));
typedef float v8f __attribute__((ext_vector_type(8)));

#define F_IN 512
#define NE   64
#define TOPK 8
#define MT   64      // tokens per block
#define KC   128     // k-chunk width (floats)
#define XP   132     // LDS chunk row pitch (floats), padded for bank-conflict-free frags
#define LP   68      // LDS logits row pitch (floats)

__global__ __launch_bounds__(128) void topk_router_kernel(
    const float* __restrict__ x, const float* __restrict__ W,
    float* __restrict__ out_w, int* __restrict__ out_idx)
{
    __shared__ float sx[MT * XP];   // x chunk, reused as logits buffer later
    __shared__ float sw[NE * XP];   // W chunk

    const int tid  = threadIdx.x;
    const int lane = tid & 31;
    const int wid  = tid >> 5;                 // 0..3
    const long tokBase = (long)blockIdx.x * MT;

    const int l15  = lane & 15;
    const int half = lane >> 4;                // 0 or 1
    const int chi  = half * 2;                 // K sub-offset for A/B frags
    const int trow = wid * 16 + l15;           // this lane's A-matrix row (token in tile)

    v8f acc0 = {}, acc1 = {}, acc2 = {}, acc3 = {};

    for (int kc = 0; kc < F_IN / KC; ++kc) {
        __syncthreads();   // previous chunk fully consumed
        // Cooperative, coalesced load of x tile and W tile (each 64 rows x 128 cols)
        #pragma unroll
        for (int it = 0; it < 16; ++it) {
            const int i   = it * 128 + tid;    // 0..2047
            const int row = i >> 5;
            const int c4  = (i & 31) * 4;
            const float4 xv = *(const float4*)(x + (tokBase + row) * F_IN + kc * KC + c4);
            *(float4*)(sx + row * XP + c4) = xv;
            const float4 wv = *(const float4*)(W + (long)row * F_IN + kc * KC + c4);
            *(float4*)(sw + row * XP + c4) = wv;
        }
        __syncthreads();

        // 32 k-steps of 4; each step: 1 A-frag + 4 B-frags + 4 WMMAs
        for (int kk = 0; kk < KC / 4; ++kk) {
            const int co = kk * 4 + chi;
            v2f a  = *(const v2f*)(sx + trow * XP + co);
            v2f b0 = *(const v2f*)(sw + ( 0 + l15) * XP + co);
            v2f b1 = *(const v2f*)(sw + (16 + l15) * XP + co);
            v2f b2 = *(const v2f*)(sw + (32 + l15) * XP + co);
            v2f b3 = *(const v2f*)(sw + (48 + l15) * XP + co);
            acc0 = __builtin_amdgcn_wmma_f32_16x16x4_f32(false, a, false, b0, (short)0, acc0, false, false);
            acc1 = __builtin_amdgcn_wmma_f32_16x16x4_f32(false, a, false, b1, (short)0, acc1, false, false);
            acc2 = __builtin_amdgcn_wmma_f32_16x16x4_f32(false, a, false, b2, (short)0, acc2, false, false);
            acc3 = __builtin_amdgcn_wmma_f32_16x16x4_f32(false, a, false, b3, (short)0, acc3, false, false);
        }
    }

    __syncthreads();          // all waves done reading sx as x-chunk; reuse as logits
    float* slog = sx;         // 64 tokens x pitch 68 = 4352 floats (< 8448)
    {
        const int wt = wid * 16;
        #pragma unroll
        for (int r = 0; r < 8; ++r) {
            const int t = wt + r + half * 8;   // C/D layout: VGPR r -> M=r (lanes 0-15), M=r+8 (16-31)
            slog[t * LP +  0 + l15] = acc0[r];
            slog[t * LP + 16 + l15] = acc1[r];
            slog[t * LP + 32 + l15] = acc2[r];
            slog[t * LP + 48 + l15] = acc3[r];
        }
    }
    __syncthreads();

    // Top-k + softmax + scatter: one lane per token (lanes 0-15 of each wave)
    if (lane < 16) {
        const int  tl    = wid * 16 + lane;
        const long token = tokBase + tl;

        float v[NE];
        #pragma unroll
        for (int e4 = 0; e4 < 16; ++e4) {
            float4 t = *(const float4*)(slog + tl * LP + e4 * 4);
            v[e4*4+0] = t.x; v[e4*4+1] = t.y; v[e4*4+2] = t.z; v[e4*4+3] = t.w;
        }

        int   bi[TOPK];
        float bv[TOPK];
        #pragma unroll
        for (int s = 0; s < TOPK; ++s) {
            float m = v[0]; int mi = 0;
            #pragma unroll
            for (int e = 1; e < NE; ++e) { if (v[e] > m) { m = v[e]; mi = e; } }
            bv[s] = m; bi[s] = mi;
            #pragma unroll
            for (int e = 0; e < NE; ++e) v[e] = (e == mi) ? -__builtin_inff() : v[e];
        }

        float p[TOPK]; float sum = 0.f;
        #pragma unroll
        for (int s = 0; s < TOPK; ++s) { p[s] = __expf(bv[s] - bv[0]); sum += p[s]; }
        const float inv = 1.f / sum;
        #pragma unroll
        for (int s = 0; s < TOPK; ++s) p[s] *= inv;

        float row[NE];
        #pragma unroll
        for (int e = 0; e < NE; ++e) {
            float o = 0.f;
            #pragma unroll
            for (int s = 0; s < TOPK; ++s) o = (bi[s] == e) ? p[s] : o;
            row[e] = o;
        }

        float* wout = out_w + token * NE;
        #pragma unroll
        for (int e4 = 0; e4 < 16; ++e4) {
            float4 t; t.x = row[e4*4+0]; t.y = row[e4*4+1]; t.z = row[e4*4+2]; t.w = row[e4*4+3];
            *(float4*)(wout + e4 * 4) = t;
        }
        int* iout = out_idx + token * TOPK;
        int4 i0; i0.x = bi[0]; i0.y = bi[1]; i0.z = bi[2]; i0.w = bi[3];
        int4 i1; i1.x = bi[4]; i1.y = bi[5]; i1.z = bi[6]; i1.w = bi[7];
        *(int4*)(iout + 0) = i0;
        *(int4*)(iout + 4) = i1;
    }
}

extern "C" void kernel_launch(void* const* d_in, const int* in_sizes, int n_in,
                              void* d_out, int out_size, void* d_ws, size_t ws_size,
                              hipStream_t stream) {
    const float* x = (const float*)d_in[0];
    const float* W = (const float*)d_in[1];
    const int B = in_sizes[0] / F_IN;                     // 262144
    float* out_w  = (float*)d_out;                        // (B, 64) f32
    int*  out_idx = (int*)((float*)d_out + (size_t)B*NE); // (B, 8) i32, concatenated after
    const int blocks = B / MT;                            // 4096
    hipLaunchKernelGGL(topk_router_kernel, dim3(blocks), dim3(128), 0, stream,
                       x, W, out_w, out_idx);
}